// CRF_34282428956936
// MI455X (gfx1250) — compile-verified
//
#include <hip/hip_runtime.h>
#include <stdint.h>

#define NL     64            // labels
#define NBATCH 32
#define SLEN   256
#define TELEM  (NL * NL)     // 4096 floats / tile = 16 KB
#define NBUF   8             // LDS ring depth (128 KB of the 320 KB WGP LDS)
#define NTHR   512           // 16 waves (wave32)
#define NWAVE  (NTHR / 32)
#define PEND   ((NBUF - 1) * 2)   // async chunks allowed outstanding (2 per tile per wave)

typedef int v4i __attribute__((vector_size(16)));
typedef __attribute__((address_space(1))) v4i gv4i_t;   // global (prints as __device__)
typedef __attribute__((address_space(3))) v4i lv4i_t;   // LDS

// 16B async global->LDS copy (CDNA5 GLOBAL_LOAD_ASYNC_TO_LDS_B128, ASYNCcnt)
__device__ __forceinline__ void async_ld16(const void* g, void* l) {
#if __has_builtin(__builtin_amdgcn_global_load_async_to_lds_b128)
  __builtin_amdgcn_global_load_async_to_lds_b128(
      (gv4i_t*)(uintptr_t)g,
      (lv4i_t*)(uint32_t)(uintptr_t)l,
      0, 0);
#else
  asm volatile("global_load_async_to_lds_b128 %0, %1, off"
               :: "v"((uint32_t)(uintptr_t)l),
                  "v"((uint64_t)(uintptr_t)g)
               : "memory");
#endif
}

#if __has_builtin(__builtin_amdgcn_s_wait_asynccnt)
#define WAIT_ASYNC(n) __builtin_amdgcn_s_wait_asynccnt(n)
#else
#define WAIT_ASYNC(n) asm volatile("s_wait_asynccnt %0" :: "i"(n) : "memory")
#endif

__device__ __forceinline__ float wred_max(float v) {
#pragma unroll
  for (int d = 16; d; d >>= 1) v = fmaxf(v, __shfl_xor(v, d, 32));
  return v;
}
__device__ __forceinline__ float wred_sum(float v) {
#pragma unroll
  for (int d = 16; d; d >>= 1) v += __shfl_xor(v, d, 32);
  return v;
}

extern "C" __global__ void __launch_bounds__(NTHR)
crf_forward(const float* __restrict__ emits,
            const long long* __restrict__ targets,
            const unsigned char* __restrict__ mask,
            float* __restrict__ ws)
{
  extern __shared__ float smem[];
  float* tiles   = smem;                     // NBUF * TELEM  (ring of emit tiles)
  float* alpha   = tiles + NBUF * TELEM;     // NL
  float* partial = alpha + NL;               // 8 * NL
  float* red     = partial + 8 * NL;         // 64 scratch floats

  const int b    = blockIdx.x;
  const int tid  = threadIdx.x;
  const int wave = tid >> 5;
  const int lane = tid & 31;
  const int j    = tid & (NL - 1);
  const int q    = tid >> 6;                 // 0..7 : i-range [8q, 8q+8)
  const float* eb = emits + (size_t)b * SLEN * TELEM;
  const int woff = wave * 1024 + lane * 16;  // this lane's byte slice of a tile

  // ---- prologue: async-prefetch tiles 1..NBUF into the LDS ring ----
  for (int t = 1; t <= NBUF; ++t) {
    const char* g = (const char*)(eb + (size_t)t * TELEM) + woff;
    char*       l = (char*)(tiles + (t & (NBUF - 1)) * TELEM) + woff;
    async_ld16(g,       l);
    async_ld16(g + 512, l + 512);
  }

  // ---- alpha_0 = emits[b, 0, BOS=0, :] ----
  if (tid < NL) {
    float a = eb[j];                         // row BOS_INDEX = 0 of tile 0
    alpha[j] = a;
    float m = wred_max(a);
    if (lane == 0) red[wave] = m;            // waves 0/1 -> red[0], red[1]
  }

  // ---- recurrence t = 1..S-1 ----
  for (int t = 1; t < SLEN; ++t) {
    WAIT_ASYNC(PEND);                        // my chunks of tile t have landed
    __syncthreads();                         // B1: tile t + alpha + maxima visible
    const float  M  = fmaxf(red[0], red[1]); // stabilizer = max_i alpha_i
    const float* tl = tiles + (t & (NBUF - 1)) * TELEM;

    float acc = 0.f;
#pragma unroll
    for (int k = 0; k < 8; ++k) {
      const int i = 8 * q + k;
      acc += __expf(alpha[i] - M + tl[i * NL + j]);   // v_exp_f32 (trans)
    }
    partial[q * NL + j] = acc;
    __syncthreads();                         // B2: partials ready, tile t consumed

    // async-prefetch tile t+NBUF into the slot just freed (clamped past the end)
    {
      int tp = t + NBUF; if (tp > SLEN - 1) tp = SLEN - 1;
      const char* g = (const char*)(eb + (size_t)tp * TELEM) + woff;
      char*       l = (char*)(tiles + ((t + NBUF) & (NBUF - 1)) * TELEM) + woff;
      async_ld16(g,       l);
      async_ld16(g + 512, l + 512);
    }

    if (tid < NL) {
      float s = 0.f;
#pragma unroll
      for (int qq = 0; qq < 8; ++qq) s += partial[qq * NL + j];
      float a_new = M + __logf(s);
      float a     = mask[b * SLEN + t] ? a_new : alpha[j];
      alpha[j] = a;
      float m = wred_max(a);
      if (lane == 0) red[wave] = m;          // next step's stabilizer pieces
    }
  }

  __syncthreads();
  // ---- log Z_b = logsumexp_j(alpha_j) ----
  if (tid < NL) {
    const float M = fmaxf(red[0], red[1]);
    float e = wred_sum(__expf(alpha[j] - M));
    if (lane == 0) red[4 + wave] = e;        // red[4], red[5]
  }
  // ---- gold-path score + token count (t = tid) ----
  float sc = 0.f, tok = 0.f;
  if (tid < SLEN) {
    const int t = tid;
    if (mask[b * SLEN + t]) {
      const long long t0 = targets[(size_t)b * (SLEN + 1) + t];
      const long long t1 = targets[(size_t)b * (SLEN + 1) + t + 1];
      sc  = eb[(size_t)t * TELEM + (size_t)t0 * NL + (size_t)t1];
      tok = 1.f;
    }
  }
  sc  = wred_sum(sc);
  tok = wred_sum(tok);
  if (lane == 0) { red[8 + wave] = sc; red[8 + NWAVE + wave] = tok; }
  __syncthreads();

  if (tid == 0) {
    const float M = fmaxf(red[0], red[1]);
    float logz = M + __logf(red[4] + red[5]);
    float scb = 0.f, tkb = 0.f;
    for (int w = 0; w < NWAVE; ++w) { scb += red[8 + w]; tkb += red[8 + NWAVE + w]; }
    ws[b]          = logz - scb;             // per-batch (logZ - score)
    ws[NBATCH + b] = tkb;                    // per-batch token count
  }
}

extern "C" __global__ void crf_finalize(const float* __restrict__ ws,
                                        float* __restrict__ out)
{
  const int lane = threadIdx.x;              // 32 threads = 1 wave
  float d  = ws[lane];
  float tk = ws[NBATCH + lane];
  d  = wred_sum(d);
  tk = wred_sum(tk);
  if (lane == 0) out[0] = d / tk;
}

extern "C" void kernel_launch(void* const* d_in, const int* in_sizes, int n_in,
                              void* d_out, int out_size, void* d_ws, size_t ws_size,
                              hipStream_t stream)
{
  (void)in_sizes; (void)n_in; (void)out_size; (void)ws_size;
  const float*         emits   = (const float*)d_in[0];
  const long long*     targets = (const long long*)d_in[1];
  const unsigned char* maskp   = (const unsigned char*)d_in[2];
  float* ws  = (float*)d_ws;
  float* out = (float*)d_out;

  const size_t shmem = (size_t)(NBUF * TELEM + NL + 8 * NL + 64) * sizeof(float);
  crf_forward<<<NBATCH, NTHR, shmem, stream>>>(emits, targets, maskp, ws);
  crf_finalize<<<1, 32, 0, stream>>>(ws, out);
}